// RetentionHead_4440996184659
// MI455X (gfx1250) — compile-verified
//
#include <hip/hip_runtime.h>

typedef __attribute__((ext_vector_type(2))) float v2f;
typedef __attribute__((ext_vector_type(8))) float v8f;

#define LOG2_GAMMA (-0.045803589825219324f) /* log2(0.96875) */

__device__ __forceinline__ float gpow(float e) {
    // gamma^e = 2^(e*log2(gamma))
    return exp2f(e * LOG2_GAMMA);
}

// Issue async DMA of one contiguous 16x64 f32 tile (4KB) from global into LDS
// with row pitch 68 floats (272B = 17*16 -> B128-aligned rows, bank stride 4).
// 256 x 16B segments, each thread issues 2 -> 4KB total; per-wave disjoint.
__device__ __forceinline__ void issue_tile_async(const float* __restrict__ gsrc,
                                                 unsigned ldsbase, int tid) {
    #pragma unroll
    for (int i = 0; i < 2; ++i) {
        int s = tid + i * 128;              // segment id 0..255
        int row = s >> 4, c16 = s & 15;     // 16 segments of 16B per row
        unsigned loff = ldsbase + (unsigned)(row * 272 + c16 * 16);
        unsigned goff = (unsigned)(s * 16); // rows contiguous in global
        asm volatile("global_load_async_to_lds_b128 %0, %1, %2"
                     :: "v"(loff), "v"(goff), "s"(gsrc) : "memory");
    }
}

__device__ __forceinline__ void wait_async0() {
    asm volatile("s_wait_asynccnt 0x0" ::: "memory");
}

// One workgroup = 4 waves (128 threads) handles one (batch, 64-row tile).
// Wave w owns rows [I64*64 + 16w, +16). Double-buffered async (memory->LDS)
// staging of 16-wide K/V column tiles overlaps the WMMA stream.
__launch_bounds__(128, 1)
__global__ void retention_wmma_f32(const float* __restrict__ q,
                                   const float* __restrict__ k,
                                   const float* __restrict__ v,
                                   float* __restrict__ out,
                                   int B, int T) {
    const int D  = 64;
    const int KP = 68;   // LDS pitch (floats): 272B rows, 16B aligned, bank stride 4
    const int SP = 18;   // pitch for per-wave S scratch

    __shared__ __attribute__((aligned(16))) float ldsK2[2][16 * 68];
    __shared__ __attribute__((aligned(16))) float ldsV2[2][16 * 68];
    __shared__ float ldsS[4][16 * 18];

    const int tilesPerB = T / 64;
    const int b    = blockIdx.x / tilesPerB;
    const int I64  = blockIdx.x % tilesPerB;

    const int tid  = threadIdx.x;
    const int wave = tid >> 5;
    const int lane = tid & 31;
    const int half = lane >> 4;   // 0: lanes 0-15, 1: lanes 16-31
    const int l16  = lane & 15;

    const int I16 = I64 * 4 + wave;                 // this wave's 16-row tile
    const long bbase = (long)b * T * D;

    // gamma^(16*192) ~ 1e-43: beyond fp32 significance vs near terms -> cut.
    const int WIN = 192;
    int jstart = 4 * I64 - WIN; if (jstart < 0) jstart = 0;
    const int jend = 4 * I64 + 3;

    const unsigned kb0 = (unsigned)(size_t)(void*)&ldsK2[0][0];
    const unsigned kb1 = (unsigned)(size_t)(void*)&ldsK2[1][0];
    const unsigned vb0 = (unsigned)(size_t)(void*)&ldsV2[0][0];
    const unsigned vb1 = (unsigned)(size_t)(void*)&ldsV2[1][0];

    // ---- prologue: start DMA of the first K/V tile into buffer 0 ----
    issue_tile_async(k + bbase + (long)jstart * 16 * D, kb0, tid);
    issue_tile_async(v + bbase + (long)jstart * 16 * D, vb0, tid);

    // --- Load Q stripe (16x64) into WMMA A-layout registers, scaled by gamma^{i_loc}.
    // A-matrix 16x4 layout: M = lane%16, K = vgpr + 2*(lane/16).
    v2f aQ[16];
    {
        const float gs = gpow((float)l16);          // gamma^{i_loc}
        const float* qrow = q + bbase + (long)I16 * 16 * D + (long)l16 * D;
        #pragma unroll
        for (int kk = 0; kk < 16; ++kk) {
            const float* p = qrow + kk * 4 + 2 * half;
            v2f t; t.x = p[0] * gs; t.y = p[1] * gs;
            aQ[kk] = t;
        }
    }

    // Output accumulators: 4 tiles of 16x16 (C/D layout: M = e + 8*half, N = l16).
    v8f acc[4];
    #pragma unroll
    for (int nb = 0; nb < 4; ++nb) acc[nb] = (v8f){0,0,0,0,0,0,0,0};

    #pragma unroll 1
    for (int j16 = jstart; j16 <= jend; ++j16) {
        const int cur = (j16 - jstart) & 1;
        const float* ldsK = ldsK2[cur];
        const float* ldsV = ldsV2[cur];

        // Tile j landed (own async writes) ...
        wait_async0();
        // ... and for everyone; also: everyone's reads of the other buffer
        // (iteration j-1) completed before they reached this barrier.
        __syncthreads();

        // Kick off DMA of tile j+1 into the other buffer; overlaps the WMMAs.
        if (j16 < jend) {
            issue_tile_async(k + bbase + (long)(j16 + 1) * 16 * D, cur ? kb0 : kb1, tid);
            issue_tile_async(v + bbase + (long)(j16 + 1) * 16 * D, cur ? vb0 : vb1, tid);
        }

        const int delta = I16 - j16;                // wave-uniform -> EXEC stays full
        if (delta >= 0 && delta <= WIN) {
            // ---- S = Qtilde * K^T  (16x16, contraction over D=64) ----
            v8f s = (v8f){0,0,0,0,0,0,0,0};
            #pragma unroll
            for (int kk = 0; kk < 16; ++kk) {
                // B-matrix 4x16: N = lane%16 (=j_loc), K(d) = vgpr + 2*half
                const float* bp = &ldsK[l16 * KP + kk * 4 + 2 * half];
                v2f bv; bv.x = bp[0]; bv.y = bp[1];
                s = __builtin_amdgcn_wmma_f32_16x16x4_f32(
                        false, aQ[kk], false, bv, (short)0, s, false, false);
            }

            // ---- decay: Q carried gamma^M; apply 0.125*gamma^(16*delta - N),
            //      N = l16 (per-lane uniform). Mask strict upper of diag tile.
            const float scale = 0.125f * gpow((float)(16 * delta - l16));
            #pragma unroll
            for (int e = 0; e < 8; ++e) {
                float val = s[e] * scale;
                if (delta == 0) {
                    int M = e + 8 * half;
                    if (M < l16) val = 0.0f;
                }
                s[e] = val;
            }

            // ---- re-layout S (C-layout regs) -> A-layout via per-wave LDS ----
            float* Sw = ldsS[wave];
            #pragma unroll
            for (int e = 0; e < 8; ++e)
                Sw[(e + 8 * half) * SP + l16] = s[e];

            // ---- out += S @ V  (16x16 x 16x64) ----
            #pragma unroll
            for (int kk = 0; kk < 4; ++kk) {
                // A slice: M = l16, K = 4kk + 2*half + {0,1}
                const float* ap = &Sw[l16 * SP + kk * 4 + 2 * half];
                v2f av; av.x = ap[0]; av.y = ap[1];
                #pragma unroll
                for (int nb = 0; nb < 4; ++nb) {
                    // B slice of V: K(j_loc) = 4kk + vgpr + 2*half, N = 16nb + l16
                    v2f bv;
                    bv.x = ldsV[(kk * 4 + 0 + 2 * half) * KP + nb * 16 + l16];
                    bv.y = ldsV[(kk * 4 + 1 + 2 * half) * KP + nb * 16 + l16];
                    acc[nb] = __builtin_amdgcn_wmma_f32_16x16x4_f32(
                                  false, av, false, bv, (short)0, acc[nb], false, false);
                }
            }
        }
    }

    // ---- write out: C/D layout M = e + 8*half, N = 16nb + l16 ----
    const long obase = bbase + (long)I16 * 16 * D;
    #pragma unroll
    for (int nb = 0; nb < 4; ++nb) {
        #pragma unroll
        for (int e = 0; e < 8; ++e) {
            int M = e + 8 * half;
            out[obase + (long)M * D + nb * 16 + l16] = acc[nb][e];
        }
    }
}

extern "C" void kernel_launch(void* const* d_in, const int* in_sizes, int n_in,
                              void* d_out, int out_size, void* d_ws, size_t ws_size,
                              hipStream_t stream) {
    (void)n_in; (void)out_size; (void)d_ws; (void)ws_size;
    const float* q = (const float*)d_in[0];
    const float* k = (const float*)d_in[1];
    const float* v = (const float*)d_in[2];
    float* out = (float*)d_out;

    const int D = 64;
    const int T = 4096;
    int B = in_sizes[0] / (T * D);
    if (B < 1) B = 1;

    dim3 grid(B * (T / 64));
    dim3 block(128);
    hipLaunchKernelGGL(retention_wmma_f32, grid, block, 0, stream, q, k, v, out, B, T);
}